// BackwardInjectDepthwiseConv2D_63617055588732
// MI455X (gfx1250) — compile-verified
//
#include <hip/hip_runtime.h>
#include <stdint.h>

// Problem constants (from reference): B=32, H=W=56, C=256, K=3, PAD=1
#define BB 32
#define HH 56
#define WW 56
#define CC 256
#define CCH 64                         // channels per block chunk
#define NCHUNK (CC / CCH)              // 4
#define ROWS 8                         // output rows per block
#define RGROUPS (HH / ROWS)            // 7
#define NBLK (BB * RGROUPS * NCHUNK)   // 896
#define ROWSTRIDE (WW * CC)            // 14336 floats per (b,h) row
#define LDS_COLS 58                    // W + 2 halo columns (zero-padded)
#define SLOT_FLOATS (LDS_COLS * CCH)   // 3712 floats per staged row
#define SMEM_FLOATS (3 * SLOT_FLOATS)  // 11136 floats (43.5 KB)
#define WS_PER_BLOCK 640               // (9 taps + bias) * 64 channels
#define NREDBLK (BB * RGROUPS)         // 224 (b,row-group) pairs per channel chunk

constexpr size_t GRAD_IN_ELEMS = (size_t)BB * HH * WW * CC; // 25,690,112

typedef float v4f __attribute__((ext_vector_type(4)));

__device__ __forceinline__ float4 fma4(float4 a, float4 b, float4 c) {
  return make_float4(fmaf(a.x, b.x, c.x), fmaf(a.y, b.y, c.y),
                     fmaf(a.z, b.z, c.z), fmaf(a.w, b.w, c.w));
}
__device__ __forceinline__ float4 add4(float4 a, float4 b) {
  return make_float4(a.x + b.x, a.y + b.y, a.z + b.z, a.w + b.w);
}
// Non-temporal 128-bit store via clang native vector (HIP float4 is a class type,
// which __builtin_nontemporal_store rejects).
__device__ __forceinline__ void nt_store4(const float4& v, float* p) {
  v4f x = { v.x, v.y, v.z, v.w };
  __builtin_nontemporal_store(x, reinterpret_cast<v4f*>(p));
}

// Fused main kernel:
//   grad_in[b,y,x,c]   = sum_{dy,dx} g_out[b,y+dy,x+dx,c] * kernels[1-dy,1-dx,c]
//   grad_wt[i,j,c]    += inputs[b,y+i-1,x+j-1,c] * g_out[b,y,x,c]   (block partials -> ws)
//   grad_bias[c]      += g_out[b,y,x,c]                              (block partials -> ws)
// g_out rows are staged into LDS with CDNA5 async global->LDS loads (ASYNCcnt),
// one row ahead of compute (s_wait_asynccnt 7 leaves the prefetched row in flight).
__global__ __launch_bounds__(256) void dwbwd_main(
    const float* __restrict__ g_out, const float* __restrict__ inputs,
    const float* __restrict__ kernels, float* __restrict__ out,
    float* __restrict__ ws) {
  __shared__ __align__(16) float smem[SMEM_FLOATS];

  const int t  = threadIdx.x;
  const int cg = t & 15;   // channel group of 4 -> 64 channels
  const int wq = t >> 4;   // 0..15 w-phase
  const int cgo = cg * 4;

  int blk = blockIdx.x;
  const int cc = blk & 3;  blk >>= 2;
  const int rg = blk % RGROUPS;
  const int b  = blk / RGROUPS;
  const int y0 = rg * ROWS;
  const int c0 = cc * CCH;

  // LDS byte address of smem base (generic shared ptr low 32 bits == LDS offset)
  const uint32_t smem_base = (uint32_t)(uintptr_t)(&smem[0]);

  // Zero the halo columns (col 0 and col 57) of all 3 row slots, once.
  for (int idx = t; idx < 3 * 2 * CCH; idx += 256) {
    const int slot = idx / (2 * CCH);
    const int rem  = idx % (2 * CCH);
    const int col  = (rem < CCH) ? 0 : (LDS_COLS - 1);
    smem[slot * SLOT_FLOATS + col * CCH + (rem & (CCH - 1))] = 0.0f;
  }

  // Zero-fill a whole staged row (for out-of-range rows): 1792 float2 units / 256 thr = 7 each
  auto zero_row = [&](int slot) {
#pragma unroll
    for (int k = 0; k < 7; ++k) {
      const int u = t + (k << 8);
      const int w = u >> 5, ui = u & 31;
      *(float2*)&smem[slot * SLOT_FLOATS + (w + 1) * CCH + ui * 2] =
          make_float2(0.0f, 0.0f);
    }
  };

  // Stage g_out row r (64-channel chunk) into LDS slot (r+1)%3 with async-to-LDS.
  // Exactly 7 async b64 per thread (uniform per wave) when the row is valid.
  auto stage_row = [&](int r) {
    const int slot = (r + 1) % 3;
    if ((unsigned)r < (unsigned)HH) {
      const float* gbase = g_out + (size_t)(b * HH + r) * ROWSTRIDE + c0;
#pragma unroll
      for (int k = 0; k < 7; ++k) {
        const int u = t + (k << 8);
        const int w = u >> 5, ui = u & 31;
        const float* gp = gbase + w * CC + ui * 2;
        const uint32_t l =
            smem_base + (uint32_t)((slot * SLOT_FLOATS + (w + 1) * CCH + ui * 2) * 4);
        asm volatile("global_load_async_to_lds_b64 %0, %1, off"
                     :: "v"(l), "v"(gp) : "memory");
      }
    } else {
      zero_row(slot);
    }
  };

  // Per-thread tap weights for grad_in: wk[i*3+j] = kernels[i,j, c0+cgo .. +3]
  float4 wk[9];
#pragma unroll
  for (int ii = 0; ii < 9; ++ii)
    wk[ii] = *(const float4*)(kernels + ii * CC + c0 + cgo);

  float4 accW[9];
#pragma unroll
  for (int ii = 0; ii < 9; ++ii) accW[ii] = make_float4(0, 0, 0, 0);
  float4 accB = make_float4(0, 0, 0, 0);

  // Prologue: stage rows y0-1 and y0
  stage_row(y0 - 1);
  stage_row(y0);

  for (int yy = 0; yy < ROWS; ++yy) {
    const int y = y0 + yy;
    const int rnext = y + 1;
    stage_row(rnext);  // prefetch next row (slot (y+2)%3, freed by last iteration)
    if ((unsigned)rnext < (unsigned)HH) {
      // rows <= y complete; the 7 just-issued (row y+1) may remain in flight
      asm volatile("s_wait_asynccnt 0x7" ::: "memory");
    } else {
      asm volatile("s_wait_asynccnt 0x0" ::: "memory");
    }
    __syncthreads();

    const int srow[3] = { ((y) % 3) * SLOT_FLOATS,        // row y-1 -> slot (y)%3
                          ((y + 1) % 3) * SLOT_FLOATS,    // row y   -> slot (y+1)%3
                          ((y + 2) % 3) * SLOT_FLOATS };  // row y+1 -> slot (y+2)%3
    const size_t orow = (size_t)(b * HH + y) * ROWSTRIDE + c0 + cgo;

    auto compute_w = [&](int w) {
      const float4 g = *(const float4*)&smem[srow[1] + (w + 1) * CCH + cgo];
      accB = add4(accB, g);
      float4 gin = make_float4(0, 0, 0, 0);
#pragma unroll
      for (int dy = -1; dy <= 1; ++dy) {
        const int sb = srow[dy + 1];
        const int ry = y + dy;
        const bool rok = ((unsigned)ry < (unsigned)HH);
        const float* irow = inputs + (size_t)(b * HH + ry) * ROWSTRIDE + c0 + cgo;
#pragma unroll
        for (int dx = -1; dx <= 1; ++dx) {
          const float4 gn = *(const float4*)&smem[sb + (w + dx + 1) * CCH + cgo];
          gin = fma4(gn, wk[4 - 3 * dy - dx], gin);  // kernels[1-dy][1-dx]
          const int rx = w + dx;
          float4 xn = make_float4(0, 0, 0, 0);
          if (rok && (unsigned)rx < (unsigned)WW)
            xn = *(const float4*)(irow + (size_t)rx * CC);
          accW[(dy + 1) * 3 + (dx + 1)] = fma4(xn, g, accW[(dy + 1) * 3 + (dx + 1)]);
        }
      }
      // grad_in is streamed out, never re-read: NT store keeps L2 for g_out/inputs
      nt_store4(gin, out + orow + (size_t)w * CC);
    };

#pragma unroll
    for (int wi = 0; wi < 3; ++wi) compute_w(wq + (wi << 4));
    if (wq < 8) compute_w(wq + 48);  // W=56 tail

    __syncthreads();  // protect slot about to be overwritten next iteration
  }

  // Block reduction of grad_wt/grad_bias partials, reusing smem (compute is done).
#pragma unroll
  for (int s = 0; s < 9; ++s)
    *(float4*)&smem[s * 1024 + wq * 64 + cgo] = accW[s];
  *(float4*)&smem[9 * 1024 + wq * 64 + cgo] = accB;
  __syncthreads();

  if (t < 160) {                 // 10 slots * 16 channel groups
    const int slot = t >> 4;
    const int cg2  = t & 15;
    float4 sum = make_float4(0, 0, 0, 0);
#pragma unroll
    for (int q = 0; q < 16; ++q)  // fixed order -> deterministic
      sum = add4(sum, *(const float4*)&smem[slot * 1024 + q * 64 + cg2 * 4]);
    nt_store4(sum, &ws[(size_t)blockIdx.x * WS_PER_BLOCK + slot * 64 + cg2 * 4]);
  }
}

// Deterministic final reduction: one block per output scalar (9*256 + 256 = 2560),
// fixed strided partition over 224 contributing blocks + fixed-order LDS tree.
__global__ __launch_bounds__(64) void dwbwd_reduce(const float* __restrict__ ws,
                                                   float* __restrict__ out) {
  __shared__ float red[64];
  const int v  = blockIdx.x;   // 0..2559
  const int s  = v >> 8;       // 0..9 (9 == bias)
  const int c  = v & 255;
  const int cc = c >> 6, ci = c & 63;
  const int t  = threadIdx.x;
  float sum = 0.0f;
  for (int q = t; q < NREDBLK; q += 64)
    sum += __builtin_nontemporal_load(
        &ws[(size_t)(q * NCHUNK + cc) * WS_PER_BLOCK + s * 64 + ci]);
  red[t] = sum;
  __syncthreads();
#pragma unroll
  for (int off = 32; off > 0; off >>= 1) {
    if (t < off) red[t] += red[t + off];
    __syncthreads();
  }
  if (t == 0) {
    const size_t o = (s < 9) ? (GRAD_IN_ELEMS + (size_t)s * 256 + c)
                             : (GRAD_IN_ELEMS + 2304 + c);
    __builtin_nontemporal_store(red[0], &out[o]);
  }
}

extern "C" void kernel_launch(void* const* d_in, const int* in_sizes, int n_in,
                              void* d_out, int out_size, void* d_ws, size_t ws_size,
                              hipStream_t stream) {
  (void)in_sizes; (void)n_in; (void)out_size; (void)ws_size;
  const float* g_out   = (const float*)d_in[0];
  const float* inputs  = (const float*)d_in[1];
  const float* kernels = (const float*)d_in[2];
  float* out = (float*)d_out;
  float* ws  = (float*)d_ws;  // needs 896 * 640 * 4 = 2,293,760 bytes

  dwbwd_main<<<NBLK, 256, 0, stream>>>(g_out, inputs, kernels, out, ws);
  dwbwd_reduce<<<10 * 256, 64, 0, stream>>>(ws, out);
}